// L2PppMaskAttn_84095459655776
// MI455X (gfx1250) — compile-verified
//
#include <hip/hip_runtime.h>
#include <cfloat>
#include <cmath>

#define B_   1024
#define L_   12
#define P_   100
#define LP_  8
#define D_   768
#define TOPK 5
#define NT_  7            // ceil(P/16) N-tiles
#define PPAD (NT_ * 16)   // 112

typedef __attribute__((ext_vector_type(2))) float v2f;
typedef __attribute__((ext_vector_type(8))) float v8f;

// ---------------------------------------------------------------------------
// Stage 1: per (l,p) compute invK = 1/max(||K||,eps) and
//          a_lp = dot(l2norm(K), l2norm(A)).  One wave32 per pair.
// ---------------------------------------------------------------------------
__global__ __launch_bounds__(32)
void prep_kernel(const float* __restrict__ K, const float* __restrict__ A,
                 float* __restrict__ a_lp, float* __restrict__ invK) {
    const int pair = blockIdx.x;            // l*P + p  (0..1199)
    const int lane = threadIdx.x;
    const float4* K4 = (const float4*)(K + (size_t)pair * D_);
    const float4* A4 = (const float4*)(A + (size_t)pair * D_);
    float kk = 0.f, aa = 0.f, ka = 0.f;
    for (int i = lane; i < D_ / 4; i += 32) {   // 6 float4 per lane
        float4 kv = K4[i];
        float4 av = A4[i];
        kk += kv.x * kv.x + kv.y * kv.y + kv.z * kv.z + kv.w * kv.w;
        aa += av.x * av.x + av.y * av.y + av.z * av.z + av.w * av.w;
        ka += kv.x * av.x + kv.y * av.y + kv.z * av.z + kv.w * av.w;
    }
    for (int off = 16; off >= 1; off >>= 1) {
        kk += __shfl_xor(kk, off, 32);
        aa += __shfl_xor(aa, off, 32);
        ka += __shfl_xor(ka, off, 32);
    }
    if (lane == 0) {
        const float ik = 1.f / fmaxf(sqrtf(kk), 1e-12f);
        const float ia = 1.f / fmaxf(sqrtf(aa), 1e-12f);
        invK[pair] = ik;
        a_lp[pair] = ka * ik * ia;
    }
}

// ---------------------------------------------------------------------------
// Stage 2: similarity GEMM via V_WMMA_F32_16X16X4_F32 + fused top-5.
// Block = 7 waves (224 thr). Wave w computes the 16x16 tile
// [16 queries b0..b0+15] x [prompts 16w..16w+15], K-dim D=768 in steps of 4.
// A layout (16x4 f32): lane l -> m = l&15, k = 2*(l>>4) + vgpr.
// B layout (4x16 f32): lane l -> n = l&15, k = 2*(l>>4) + vgpr.
// D layout (16x16 f32): lane l -> n = l&15, m = r + 8*(l>>4), r = vgpr 0..7.
// ---------------------------------------------------------------------------
__global__ __launch_bounds__(NT_ * 32)
void scores_topk_kernel(const float* __restrict__ x, const float* __restrict__ K,
                        const float* __restrict__ invK, const float* __restrict__ a_lp,
                        float* __restrict__ wght, int* __restrict__ idx) {
    __shared__ float smem[16 * PPAD];

    const int l    = blockIdx.y;
    const int b0   = blockIdx.x * 16;
    const int wave = threadIdx.x >> 5;
    const int lane = threadIdx.x & 31;
    const int half = lane >> 4;        // 0 or 1
    const int mn   = lane & 15;        // m for A-side, n for B-side
    const int p    = wave * 16 + mn;   // this lane's prompt column
    const int pc   = (p < P_) ? p : (P_ - 1);   // clamp for safe loads

    // x_query is [B, L, D]; K_all is [L, P, D]
    const float* ap = x + ((size_t)(b0 + mn) * L_ + l) * D_ + 2 * half;
    const float* bp = K + ((size_t)l * P_ + pc) * D_ + 2 * half;

    v8f c = {};
#pragma unroll 4
    for (int d = 0; d < D_; d += 4) {
        v2f a = *(const v2f*)ap;   // 8B aligned: k = 2*half + {0,1}
        v2f b = *(const v2f*)bp;
        c = __builtin_amdgcn_wmma_f32_16x16x4_f32(
                /*neg_a=*/false, a, /*neg_b=*/false, b,
                /*c_mod=*/(short)0, c, /*reuse_a=*/false, /*reuse_b=*/false);
        ap += 4;
        bp += 4;
    }

    // Scale by 1/||K|| (ranking-equivalent to full cosine) and stash in LDS.
    const float scl = (p < P_) ? invK[l * P_ + p] : 0.f;
#pragma unroll
    for (int r = 0; r < 8; ++r) {
        const int m = r + 8 * half;
        smem[m * PPAD + p] = (p < P_) ? (c[r] * scl) : -FLT_MAX;
    }
    __syncthreads();

    // Stable top-5 (lowest index wins ties, like jax.lax.top_k).
    if (threadIdx.x < 16) {
        const int m = threadIdx.x;
        const int b = b0 + m;
        float* row = smem + m * PPAD;
        const size_t base = ((size_t)l * B_ + b) * TOPK;
#pragma unroll
        for (int k = 0; k < TOPK; ++k) {
            float best = -FLT_MAX;
            int   bi   = 0;
            for (int q = 0; q < PPAD; ++q) {
                const float s = row[q];
                if (s > best) { best = s; bi = q; }
            }
            row[bi]        = -FLT_MAX;
            wght[base + k] = a_lp[l * P_ + bi];
            idx[base + k]  = bi;
        }
    }
}

// ---------------------------------------------------------------------------
// Stage 3: out[l,b,:,:] = sum_k w_k * P_all[l, idx_k, :, :]
// One block per (l,b): 6144-float row, float4 vectorized. Store-BW bound.
// ---------------------------------------------------------------------------
__global__ __launch_bounds__(256)
void combine_kernel(const float* __restrict__ Pall, const float* __restrict__ wght,
                    const int* __restrict__ idx, float* __restrict__ out) {
    const int b = blockIdx.x;
    const int l = blockIdx.y;
    const size_t base = ((size_t)l * B_ + b) * TOPK;

    const float w0 = wght[base + 0], w1 = wght[base + 1], w2 = wght[base + 2],
                w3 = wght[base + 3], w4 = wght[base + 4];
    const int i0 = idx[base + 0], i1 = idx[base + 1], i2 = idx[base + 2],
              i3 = idx[base + 3], i4 = idx[base + 4];

    const int RD = LP_ * D_;   // 6144 floats per (l,b) row
    const float4* p0 = (const float4*)(Pall + ((size_t)l * P_ + i0) * RD);
    const float4* p1 = (const float4*)(Pall + ((size_t)l * P_ + i1) * RD);
    const float4* p2 = (const float4*)(Pall + ((size_t)l * P_ + i2) * RD);
    const float4* p3 = (const float4*)(Pall + ((size_t)l * P_ + i3) * RD);
    const float4* p4 = (const float4*)(Pall + ((size_t)l * P_ + i4) * RD);
    float4* o = (float4*)(out + ((size_t)l * B_ + b) * RD);

    for (int i = threadIdx.x; i < RD / 4; i += 256) {   // 6 iters/thread
        const float4 a0 = p0[i], a1 = p1[i], a2 = p2[i], a3 = p3[i], a4 = p4[i];
        float4 r;
        r.x = w0 * a0.x + w1 * a1.x + w2 * a2.x + w3 * a3.x + w4 * a4.x;
        r.y = w0 * a0.y + w1 * a1.y + w2 * a2.y + w3 * a3.y + w4 * a4.y;
        r.z = w0 * a0.z + w1 * a1.z + w2 * a2.z + w3 * a3.z + w4 * a4.z;
        r.w = w0 * a0.w + w1 * a1.w + w2 * a2.w + w3 * a3.w + w4 * a4.w;
        o[i] = r;
    }
}

// ---------------------------------------------------------------------------
extern "C" void kernel_launch(void* const* d_in, const int* in_sizes, int n_in,
                              void* d_out, int out_size, void* d_ws, size_t ws_size,
                              hipStream_t stream) {
    const float* x    = (const float*)d_in[0];   // [B, L, D]
    const float* Kall = (const float*)d_in[1];   // [L, P, D]
    const float* Aall = (const float*)d_in[2];   // [L, P, D]
    const float* Pall = (const float*)d_in[3];   // [L, P, Lp, D]
    float* out = (float*)d_out;                  // [L, B, Lp, D]

    // Workspace layout (floats): a_lp[1200] | invK[1200] | wght[L*B*5] | idx[L*B*5]
    float* ws   = (float*)d_ws;
    float* a_lp = ws;
    float* invK = ws + L_ * P_;
    float* wght = ws + 2 * L_ * P_;
    int*   idx  = (int*)(ws + 2 * L_ * P_ + (size_t)L_ * B_ * TOPK);

    prep_kernel<<<L_ * P_, 32, 0, stream>>>(Kall, Aall, a_lp, invK);
    scores_topk_kernel<<<dim3(B_ / 16, L_), NT_ * 32, 0, stream>>>(
        x, Kall, invK, a_lp, wght, idx);
    combine_kernel<<<dim3(B_, L_), 256, 0, stream>>>(Pall, wght, idx, out);
}